// DGCNN_14156212208341
// MI455X (gfx1250) — compile-verified
//
#include <hip/hip_runtime.h>

// ---------------------------------------------------------------------------
// DGCNN inference for gfx1250 (MI455X), fp32 WMMA (v_wmma_f32_16x16x4_f32).
// ---------------------------------------------------------------------------

typedef float v2f __attribute__((ext_vector_type(2)));
typedef float v8f __attribute__((ext_vector_type(8)));

#define NPTS 2048
#define NBATCH 8
#define KNN 20
#define NEG_SLOPE 0.2f
#define BN_EPS 1e-5f
#define EDGE_BLK 128
#define NEG_BIG (-3.4e38f)

// ---- WMMA helper: D = A(16x4,f32) * B(4x16,f32) + C --------------------------
__device__ __forceinline__ v8f wmma4(v2f a, v2f b, v8f c) {
  // 8 args: (neg_a, A, neg_b, B, c_mod, C, reuse_a, reuse_b)
  return __builtin_amdgcn_wmma_f32_16x16x4_f32(false, a, false, b, (short)0, c,
                                               false, false);
}

__device__ __forceinline__ v8f zero8() {
  v8f z = {0.f, 0.f, 0.f, 0.f, 0.f, 0.f, 0.f, 0.f};
  return z;
}

// A-matrix 16x4 f32 fragment (ISA 7.12.2): lanes 0-15 rows 0-15 K={kk,kk+1},
// lanes 16-31 rows 0-15 K={kk+2,kk+3}.  Works for LDS or global pointers.
__device__ __forceinline__ v2f loadAfrag(const float* A, int stride, int kk) {
  int lane = threadIdx.x & 31;
  int ks = kk + ((lane >> 4) << 1);
  const float* p = A + (size_t)(lane & 15) * stride + ks;
  v2f a;
  a.x = p[0];
  a.y = p[1];
  return a;
}

// W element with edge-feature column mapping: packed e = [diff(Cpad) | ctr(Cpad)],
// real layout in W is [diff(Chalf) | ctr(Chalf)]; pad channels read as 0.
__device__ __forceinline__ float wval(const float* W, int Estride, int Chalf,
                                      int Cpad, int o, int k) {
  int half = (k >= Cpad) ? 1 : 0;
  int c = k - half * Cpad;
  return (c < Chalf) ? W[(size_t)o * Estride + half * Chalf + c] : 0.f;
}

// B-matrix 4x16 fragment from weight matrix: B[k][o] = W[o][map(k)]
__device__ __forceinline__ v2f loadBfragW(const float* W, int Estride, int Chalf,
                                          int Cpad, int obase, int kk) {
  int lane = threadIdx.x & 31;
  int o = obase + (lane & 15);
  int ks = kk + ((lane >> 4) << 1);
  v2f b;
  b.x = wval(W, Estride, Chalf, Cpad, o, ks);
  b.y = wval(W, Estride, Chalf, Cpad, o, ks + 1);
  return b;
}

// Full 16x16 output tile: A is a 16 x Epad tile (row-major, stride Astride).
__device__ __forceinline__ v8f gemm_tile(const float* A, int Astride, int Epad,
                                         const float* W, int Estride, int Chalf,
                                         int Cpad, int obase) {
  v8f acc = zero8();
  for (int kk = 0; kk < Epad; kk += 4)
    acc = wmma4(loadAfrag(A, Astride, kk),
                loadBfragW(W, Estride, Chalf, Cpad, obase, kk), acc);
  return acc;
}

// Monotone float <-> uint encoding so "max" can use unsigned atomicMax.
__device__ __forceinline__ unsigned f2ord(float f) {
  unsigned u = __float_as_uint(f);
  return (u & 0x80000000u) ? ~u : (u | 0x80000000u);
}
__device__ __forceinline__ float ord2f(unsigned u) {
  return (u & 0x80000000u) ? __uint_as_float(u & 0x7FFFFFFFu)
                           : __uint_as_float(~u);
}

// ---------------------------------------------------------------------------
// Kernels
// ---------------------------------------------------------------------------

// x (B,3,N) -> xt (B,N,4) with zero pad channel.
__global__ void k_transpose(const float* __restrict__ x, float* __restrict__ xt) {
  int t = blockIdx.x * blockDim.x + threadIdx.x;
  if (t >= NBATCH * NPTS * 4) return;
  int c = t & 3;
  int n = (t >> 2) & (NPTS - 1);
  int b = t >> 13;
  xt[t] = (c < 3) ? x[((size_t)b * 3 + c) * NPTS + n] : 0.f;
}

// squared norms per point
__global__ void k_rownorm(const float* __restrict__ src, int C, int Cs,
                          float* __restrict__ sq) {
  int t = blockIdx.x * blockDim.x + threadIdx.x;
  if (t >= NBATCH * NPTS) return;
  const float* p = src + (size_t)t * Cs;
  float s = 0.f;
  for (int c = 0; c < C; ++c) { float v = p[c]; s += v * v; }
  sq[t] = s;
}

// pd[r][c] = 2*<x_r,x_c> - |x_r|^2 - |x_c|^2 for one batch, via WMMA tiles.
__global__ void k_dist(const float* __restrict__ src, int Cs, int C,
                       const float* __restrict__ sq, int b,
                       float* __restrict__ dist) {
  const int TPC = NPTS / 16;  // 128 tiles per side
  int wave = threadIdx.x >> 5;
  int tile = blockIdx.x * (blockDim.x >> 5) + wave;
  if (tile >= TPC * TPC) return;
  int tr = tile / TPC, tc = tile - tr * TPC;
  const float* rowsA = src + ((size_t)b * NPTS + tr * 16) * Cs;
  const float* rowsB = src + ((size_t)b * NPTS + tc * 16) * Cs;
  v8f acc = zero8();
  for (int kk = 0; kk < C; kk += 4)
    acc = wmma4(loadAfrag(rowsA, Cs, kk), loadAfrag(rowsB, Cs, kk), acc);
  int lane = threadIdx.x & 31;
  int col = tc * 16 + (lane & 15);
  int rbase = tr * 16 + ((lane >> 4) << 3);
  float sc = sq[b * NPTS + col];
#pragma unroll
  for (int v = 0; v < 8; ++v) {
    int r = rbase + v;
    float pd = 2.f * acc[v] - sq[b * NPTS + r] - sc;
    dist[(size_t)r * NPTS + col] = pd;
  }
}

// Wave-per-row top-20 (largest pd).  Per-lane sorted top-20 of a strided
// subset, then a 20-round cross-lane merge with shfl reductions.
__global__ void k_topk(const float* __restrict__ dist, int* __restrict__ idxout,
                       int b) {
  int wave = threadIdx.x >> 5;
  int row = blockIdx.x * (blockDim.x >> 5) + wave;
  if (row >= NPTS) return;
  int lane = threadIdx.x & 31;
  float ld[KNN];
  int li[KNN];
#pragma unroll
  for (int i = 0; i < KNN; ++i) { ld[i] = NEG_BIG; li[i] = 0x7fffffff; }
  const float* drow = dist + (size_t)row * NPTS;
  for (int j = lane; j < NPTS; j += 32) {
    float v = drow[j];
    if (v > ld[KNN - 1]) {
      int p = KNN - 1;
      while (p > 0 && ld[p - 1] < v) {
        ld[p] = ld[p - 1];
        li[p] = li[p - 1];
        --p;
      }
      ld[p] = v;
      li[p] = j;
    }
  }
  int p = 0;
  for (int t = 0; t < KNN; ++t) {
    float cv = (p < KNN) ? ld[p] : NEG_BIG;
    int ci = (p < KNN) ? li[p] : 0x7fffffff;
    float bv = cv;
    int bi = ci;
    int bl = lane;
    for (int off = 16; off; off >>= 1) {
      float ov = __shfl_xor(bv, off, 32);
      int oi = __shfl_xor(bi, off, 32);
      int ol = __shfl_xor(bl, off, 32);
      if (ov > bv || (ov == bv && oi < bi)) { bv = ov; bi = oi; bl = ol; }
    }
    if (lane == bl) ++p;
    if (lane == 0) idxout[((size_t)b * NPTS + row) * KNN + t] = bi;
  }
}

// Build 16 x Epad edge-feature tile in LDS: e = [nbr - ctr | ctr]
__device__ __forceinline__ void build_etile(float* eT, const float* __restrict__ src,
                                            int Cs, int Cpad,
                                            const int* __restrict__ idx, int b,
                                            int rowbase, int Epad) {
  for (int i = threadIdx.x; i < 16 * Epad; i += EDGE_BLK) {
    int rl = i / Epad, j = i - rl * Epad;
    int row = rowbase + rl;
    int n = row / KNN, kk = row - n * KNN;
    int half = (j >= Cpad) ? 1 : 0;
    int c = j - half * Cpad;
    float ctr = src[((size_t)b * NPTS + n) * Cs + c];
    float val;
    if (half) {
      val = ctr;
    } else {
      int ni = idx[((size_t)b * NPTS + n) * KNN + kk];
      val = src[((size_t)b * NPTS + ni) * Cs + c] - ctr;
    }
    eT[rl * Epad + j] = val;
  }
}

// EdgeConv pass 1: per-channel sum / sumsq of pre-BN h, deterministic
// (LDS partials with unique owners -> per-block partial buffer).
__global__ void k_edge_pass1(const float* __restrict__ src, int Cs, int Cpad,
                             int Chalf, const int* __restrict__ idx,
                             const float* __restrict__ W, int O,
                             float* __restrict__ partial) {
  __shared__ float eT[16 * 256];
  __shared__ float pS[512];
  int b = blockIdx.y;
  int nb = blockIdx.x * 16;
  int Epad = 2 * Cpad;
  int tid = threadIdx.x, wave = tid >> 5, lane = tid & 31;
  for (int i = tid; i < 512; i += EDGE_BLK) pS[i] = 0.f;
  int Ot = O >> 4;
  for (int rt = 0; rt < 20; ++rt) {
    __syncthreads();
    int rowbase = nb * KNN + rt * 16;
    build_etile(eT, src, Cs, Cpad, idx, b, rowbase, Epad);
    __syncthreads();
    for (int ot = wave; ot < Ot; ot += EDGE_BLK / 32) {
      v8f acc = gemm_tile(eT, Epad, Epad, W, 2 * Chalf, Chalf, Cpad, ot * 16);
      float s = 0.f, s2 = 0.f;
#pragma unroll
      for (int v = 0; v < 8; ++v) { float h = acc[v]; s += h; s2 += h * h; }
      s += __shfl_xor(s, 16, 32);
      s2 += __shfl_xor(s2, 16, 32);
      if (lane < 16) {  // unique owner per (block, o)
        int o = ot * 16 + lane;
        pS[o] += s;
        pS[O + o] += s2;
      }
    }
  }
  __syncthreads();
  int O2 = 2 * O;
  size_t blk = (size_t)blockIdx.y * gridDim.x + blockIdx.x;
  for (int i = tid; i < O2; i += EDGE_BLK) partial[blk * O2 + i] = pS[i];
}

// EdgeConv pass 2: recompute h, apply BN scale/shift + LeakyReLU, max over k
// via ordered-uint LDS atomicMax, write (B,N,O).
__global__ void k_edge_pass2(const float* __restrict__ src, int Cs, int Cpad,
                             int Chalf, const int* __restrict__ idx,
                             const float* __restrict__ W, int O,
                             const float* __restrict__ scsh,
                             float* __restrict__ dst) {
  __shared__ float eT[16 * 256];
  __shared__ unsigned maxb[16 * 256];
  int b = blockIdx.y;
  int nb = blockIdx.x * 16;
  int Epad = 2 * Cpad;
  int tid = threadIdx.x, wave = tid >> 5, lane = tid & 31;
  for (int i = tid; i < 16 * O; i += EDGE_BLK) maxb[i] = 0u;
  int Ot = O >> 4;
  for (int rt = 0; rt < 20; ++rt) {
    __syncthreads();
    int rowbase = nb * KNN + rt * 16;
    build_etile(eT, src, Cs, Cpad, idx, b, rowbase, Epad);
    __syncthreads();
    for (int ot = wave; ot < Ot; ot += EDGE_BLK / 32) {
      v8f acc = gemm_tile(eT, Epad, Epad, W, 2 * Chalf, Chalf, Cpad, ot * 16);
      int o = ot * 16 + (lane & 15);
      float sc = scsh[o], sh = scsh[O + o];
      int mb = (lane >> 4) << 3;
#pragma unroll
      for (int v = 0; v < 8; ++v) {
        float y = acc[v] * sc + sh;
        y = (y >= 0.f) ? y : NEG_SLOPE * y;
        int nl = (rt * 16 + mb + v) / KNN;  // local point id 0..15
        atomicMax(&maxb[nl * O + o], f2ord(y));
      }
    }
  }
  __syncthreads();
  for (int i = tid; i < 16 * O; i += EDGE_BLK) {
    int nl = i / O, o = i - nl * O;
    dst[((size_t)b * NPTS + nb + nl) * O + o] = ord2f(maxb[i]);
  }
}

// Fixed-order partial reduction (deterministic BN stats).
__global__ void k_reduce_partial(const float* __restrict__ partial, int nblk,
                                 int O2, float* __restrict__ sums) {
  int t = blockIdx.x * blockDim.x + threadIdx.x;
  if (t >= O2) return;
  float s = 0.f;
  for (int i = 0; i < nblk; ++i) s += partial[(size_t)i * O2 + t];
  sums[t] = s;
}

__global__ void k_bnfin(const float* __restrict__ sums, float invcnt,
                        const float* __restrict__ g, const float* __restrict__ bb,
                        int O, float* __restrict__ scsh) {
  int o = blockIdx.x * blockDim.x + threadIdx.x;
  if (o >= O) return;
  float m = sums[o] * invcnt;
  float var = sums[O + o] * invcnt - m * m;
  float sc = g[o] * rsqrtf(var + BN_EPS);
  scsh[o] = sc;
  scsh[O + o] = bb[o] - m * sc;
}

// Point-conv pass 1: stats over all B*N rows (16 rows per block).
__global__ void k_point_pass1(const float* __restrict__ src, int E,
                              const float* __restrict__ W, int O,
                              float* __restrict__ partial) {
  __shared__ float pS[512];
  int tid = threadIdx.x, wave = tid >> 5, lane = tid & 31;
  for (int i = tid; i < 512; i += 128) pS[i] = 0.f;
  __syncthreads();
  const float* A = src + (size_t)blockIdx.x * 16 * E;
  int Ot = O >> 4;
  for (int ot = wave; ot < Ot; ot += 4) {
    v8f acc = gemm_tile(A, E, E, W, E, E, E, ot * 16);
    float s = 0.f, s2 = 0.f;
#pragma unroll
    for (int v = 0; v < 8; ++v) { float h = acc[v]; s += h; s2 += h * h; }
    s += __shfl_xor(s, 16, 32);
    s2 += __shfl_xor(s2, 16, 32);
    if (lane < 16) {
      int o = ot * 16 + lane;
      pS[o] = s;
      pS[O + o] = s2;
    }
  }
  __syncthreads();
  int O2 = 2 * O;
  for (int i = tid; i < O2; i += 128)
    partial[(size_t)blockIdx.x * O2 + i] = pS[i];
}

// Point-conv pass 2: BN + LeakyReLU; optional transposed (B,O,N) output.
__global__ void k_point_pass2(const float* __restrict__ src, int E,
                              const float* __restrict__ W, int O,
                              const float* __restrict__ scsh,
                              float* __restrict__ dst, int transposed) {
  int tid = threadIdx.x, wave = tid >> 5, lane = tid & 31;
  size_t rowbase = (size_t)blockIdx.x * 16;
  const float* A = src + rowbase * E;
  int Ot = O >> 4;
  for (int ot = wave; ot < Ot; ot += 4) {
    v8f acc = gemm_tile(A, E, E, W, E, E, E, ot * 16);
    int o = ot * 16 + (lane & 15);
    float sc = scsh[o], sh = scsh[O + o];
    int mb = (lane >> 4) << 3;
#pragma unroll
    for (int v = 0; v < 8; ++v) {
      float y = acc[v] * sc + sh;
      y = (y >= 0.f) ? y : NEG_SLOPE * y;
      size_t row = rowbase + mb + v;
      if (transposed) {
        size_t b = row >> 11;
        size_t n = row & (NPTS - 1);
        dst[(b * O + o) * NPTS + n] = y;
      } else {
        dst[row * O + o] = y;
      }
    }
  }
}

// global max over points: gmax[b][o] = max_n h6[b][n][o]
__global__ void k_colmax(const float* __restrict__ h6, float* __restrict__ gmax) {
  int t = blockIdx.x * blockDim.x + threadIdx.x;
  if (t >= NBATCH * 256) return;
  int b = t >> 8, o = t & 255;
  const float* p = h6 + (size_t)b * NPTS * 256 + o;
  float m = NEG_BIG;
  for (int n = 0; n < NPTS; ++n) m = fmaxf(m, p[(size_t)n * 256]);
  gmax[t] = m;
}

// cat = [x1(64) | x2(64) | x3(128) | x4(256) | gmax-broadcast(256)]
__global__ void k_cat(const float* __restrict__ x1, const float* __restrict__ x2,
                      const float* __restrict__ x3, const float* __restrict__ x4,
                      const float* __restrict__ gmax, float* __restrict__ cat) {
  size_t t = (size_t)blockIdx.x * blockDim.x + threadIdx.x;
  if (t >= (size_t)NBATCH * NPTS * 768) return;
  size_t row = t / 768;
  int j = (int)(t - row * 768);
  float v;
  if (j < 64) v = x1[row * 64 + j];
  else if (j < 128) v = x2[row * 64 + (j - 64)];
  else if (j < 256) v = x3[row * 128 + (j - 128)];
  else if (j < 512) v = x4[row * 256 + (j - 256)];
  else { size_t b = row >> 11; v = gmax[b * 256 + (j - 512)]; }
  cat[t] = v;
}

// ---------------------------------------------------------------------------
// Host orchestration
// ---------------------------------------------------------------------------

// workspace offsets (in floats)
static constexpr size_t OFF_XT = 0;
static constexpr size_t OFF_X1 = OFF_XT + (size_t)NBATCH * NPTS * 4;
static constexpr size_t OFF_X2 = OFF_X1 + (size_t)NBATCH * NPTS * 64;
static constexpr size_t OFF_X3 = OFF_X2 + (size_t)NBATCH * NPTS * 64;
static constexpr size_t OFF_X4 = OFF_X3 + (size_t)NBATCH * NPTS * 128;
static constexpr size_t OFF_H6 = OFF_X4 + (size_t)NBATCH * NPTS * 256;
static constexpr size_t OFF_CAT = OFF_H6 + (size_t)NBATCH * NPTS * 256;
static constexpr size_t OFF_GMAX = OFF_CAT + (size_t)NBATCH * NPTS * 768;
static constexpr size_t OFF_SQ = OFF_GMAX + (size_t)NBATCH * 256;
static constexpr size_t OFF_SUMS = OFF_SQ + (size_t)NBATCH * NPTS;
static constexpr size_t OFF_SCSH = OFF_SUMS + 512;
static constexpr size_t OFF_PART = OFF_SCSH + 512;
static constexpr size_t OFF_DIST = OFF_PART + (size_t)1024 * 512;
static constexpr size_t OFF_IDX = OFF_DIST + (size_t)NPTS * NPTS;  // ints

extern "C" void kernel_launch(void* const* d_in, const int* in_sizes, int n_in,
                              void* d_out, int out_size, void* d_ws,
                              size_t ws_size, hipStream_t stream) {
  (void)in_sizes; (void)n_in; (void)out_size; (void)ws_size;
  const float* x = (const float*)d_in[0];
  const float* W1 = (const float*)d_in[1];
  const float* g1 = (const float*)d_in[2];
  const float* b1 = (const float*)d_in[3];
  const float* W2 = (const float*)d_in[4];
  const float* g2 = (const float*)d_in[5];
  const float* b2 = (const float*)d_in[6];
  const float* W3 = (const float*)d_in[7];
  const float* g3 = (const float*)d_in[8];
  const float* b3 = (const float*)d_in[9];
  const float* W4 = (const float*)d_in[10];
  const float* g4 = (const float*)d_in[11];
  const float* b4 = (const float*)d_in[12];
  const float* W6 = (const float*)d_in[13];
  const float* g6 = (const float*)d_in[14];
  const float* b6 = (const float*)d_in[15];
  const float* W5 = (const float*)d_in[16];
  const float* g5 = (const float*)d_in[17];
  const float* b5 = (const float*)d_in[18];

  float* ws = (float*)d_ws;
  float* XT = ws + OFF_XT;
  float* X1 = ws + OFF_X1;
  float* X2 = ws + OFF_X2;
  float* X3 = ws + OFF_X3;
  float* X4 = ws + OFF_X4;
  float* H6 = ws + OFF_H6;
  float* CAT = ws + OFF_CAT;
  float* GMAX = ws + OFF_GMAX;
  float* SQ = ws + OFF_SQ;
  float* SUMS = ws + OFF_SUMS;
  float* SCSH = ws + OFF_SCSH;
  float* PART = ws + OFF_PART;
  float* DIST = ws + OFF_DIST;
  int* IDX = (int*)(ws + OFF_IDX);

  k_transpose<<<(NBATCH * NPTS * 4 + 255) / 256, 256, 0, stream>>>(x, XT);

  auto edge_layer = [&](const float* src, int Cs, int Cpad, int Chalf,
                        const float* W, const float* g, const float* bb, int O,
                        float* dst) {
    // kNN: distances (per batch, reusing DIST via stream ordering) + top-20
    k_rownorm<<<(NBATCH * NPTS + 255) / 256, 256, 0, stream>>>(src, Cpad, Cs, SQ);
    for (int b = 0; b < NBATCH; ++b) {
      k_dist<<<2048, 256, 0, stream>>>(src, Cs, Cpad, SQ, b, DIST);
      k_topk<<<NPTS / 4, 128, 0, stream>>>(DIST, IDX, b);
    }
    // two-pass GEMM + BN + LeakyReLU + max over k
    k_edge_pass1<<<dim3(NPTS / 16, NBATCH), EDGE_BLK, 0, stream>>>(
        src, Cs, Cpad, Chalf, IDX, W, O, PART);
    k_reduce_partial<<<(2 * O + 255) / 256, 256, 0, stream>>>(
        PART, (NPTS / 16) * NBATCH, 2 * O, SUMS);
    k_bnfin<<<1, 256, 0, stream>>>(SUMS,
                                   1.f / ((float)NBATCH * NPTS * KNN), g, bb, O,
                                   SCSH);
    k_edge_pass2<<<dim3(NPTS / 16, NBATCH), EDGE_BLK, 0, stream>>>(
        src, Cs, Cpad, Chalf, IDX, W, O, SCSH, dst);
  };

  edge_layer(XT, 4, 4, 3, W1, g1, b1, 64, X1);     // E=6 (padded 8)  -> 64
  edge_layer(X1, 64, 64, 64, W2, g2, b2, 64, X2);  // E=128 -> 64
  edge_layer(X2, 64, 64, 64, W3, g3, b3, 128, X3); // E=128 -> 128
  edge_layer(X3, 128, 128, 128, W4, g4, b4, 256, X4); // E=256 -> 256

  auto point_conv = [&](const float* src, int E, const float* W, const float* g,
                        const float* bb, int O, float* dst, int transposed) {
    int nblk = NBATCH * NPTS / 16;  // 1024
    k_point_pass1<<<nblk, 128, 0, stream>>>(src, E, W, O, PART);
    k_reduce_partial<<<(2 * O + 255) / 256, 256, 0, stream>>>(PART, nblk, 2 * O,
                                                              SUMS);
    k_bnfin<<<1, 256, 0, stream>>>(SUMS, 1.f / ((float)NBATCH * NPTS), g, bb, O,
                                   SCSH);
    k_point_pass2<<<nblk, 128, 0, stream>>>(src, E, W, O, SCSH, dst, transposed);
  };

  point_conv(X4, 256, W6, g6, b6, 256, H6, 0);
  k_colmax<<<NBATCH, 256, 0, stream>>>(H6, GMAX);
  k_cat<<<(int)(((size_t)NBATCH * NPTS * 768 + 255) / 256), 256, 0, stream>>>(
      X1, X2, X3, X4, GMAX, CAT);
  point_conv(CAT, 768, W5, g5, b5, 256, (float*)d_out, 1);
}